// gat_model_83777632076015
// MI455X (gfx1250) — compile-verified
//
#include <hip/hip_runtime.h>
#include <math.h>

#define NN 10000
#define NE 320000
#define NH 8
#define ND 32
#define F1 256            // NH*ND
#define NEG_SL 0.2f

typedef __attribute__((ext_vector_type(2))) float v2f;
typedef __attribute__((ext_vector_type(8))) float v8f;

// monotonic order-preserving float<->uint encoding for atomicMax-based segment max
__device__ __forceinline__ unsigned enc_f(float f) {
    unsigned u = __float_as_uint(f);
    return (u & 0x80000000u) ? ~u : (u | 0x80000000u);
}
__device__ __forceinline__ float dec_f(unsigned u) {
    unsigned v = (u & 0x80000000u) ? (u & 0x7FFFFFFFu) : ~u;
    return __uint_as_float(v);
}
#define ENC_NEG_INF 0x007FFFFFu   // enc(-inf)

__device__ __forceinline__ float lrelu(float v) { return v > 0.f ? v : NEG_SL * v; }
__device__ __forceinline__ float elu(float v)   { return v > 0.f ? v : expm1f(v); }

// --- layer-1 rank-1 attention coefficients: cs1[h] = W1[h,:]·att_src1[h,:], cd1 likewise
__global__ void k_prep(const float* __restrict__ W1, const float* __restrict__ as1,
                       const float* __restrict__ ad1, float* cs1, float* cd1) {
    int h = threadIdx.x;
    if (h < NH) {
        float s = 0.f, d = 0.f;
        for (int j = 0; j < ND; ++j) {
            float w = W1[h * ND + j];
            s += w * as1[h * ND + j];
            d += w * ad1[h * ND + j];
        }
        cs1[h] = s; cd1[h] = d;
    }
}

__global__ void k_init(unsigned* emax1, float* denom1, float* S1,
                       unsigned* emax2, float* denom2, float* agg2, float* pooled) {
    int i = blockIdx.x * blockDim.x + threadIdx.x;
    if (i < NN * NH) { emax1[i] = ENC_NEG_INF; denom1[i] = 0.f; S1[i] = 0.f; }
    if (i < NN)      { emax2[i] = ENC_NEG_INF; denom2[i] = 0.f; }
    if (i < NN * ND) agg2[i] = 0.f;
    if (i < ND)      pooled[i] = 0.f;
}

// --- layer-1 edge pass 1: segment max of leaky_relu(x[s]*cs + x[t]*cd)
__global__ void k_edge1_max(const int* __restrict__ src, const int* __restrict__ dst,
                            const float* __restrict__ x, const float* __restrict__ cs1,
                            const float* __restrict__ cd1, unsigned* emax1) {
    int e = blockIdx.x * blockDim.x + threadIdx.x;
    if (e >= NE) return;
    int s = src[e], t = dst[e];
    float xs = x[s], xt = x[t];
#pragma unroll
    for (int h = 0; h < NH; ++h) {
        float v = lrelu(xs * cs1[h] + xt * cd1[h]);
        atomicMax(&emax1[t * NH + h], enc_f(v));
    }
}

// --- layer-1 edge pass 2: denom += ex, S += ex*x[src]  (rank-1 message collapse)
__global__ void k_edge1_sum(const int* __restrict__ src, const int* __restrict__ dst,
                            const float* __restrict__ x, const float* __restrict__ cs1,
                            const float* __restrict__ cd1, const unsigned* __restrict__ emax1,
                            float* denom1, float* S1) {
    int e = blockIdx.x * blockDim.x + threadIdx.x;
    if (e >= NE) return;
    int s = src[e], t = dst[e];
    float xs = x[s], xt = x[t];
#pragma unroll
    for (int h = 0; h < NH; ++h) {
        float v  = lrelu(xs * cs1[h] + xt * cd1[h]);
        float ex = expf(v - dec_f(emax1[t * NH + h]));
        atomicAdd(&denom1[t * NH + h], ex);
        atomicAdd(&S1[t * NH + h], ex * xs);
    }
}

// --- layer-1 node features: H1[n, h*32+d] = elu(W1[c]*S1[n,h]/(denom+eps) + b1[c])
__global__ void k_h1(const float* __restrict__ W1, const float* __restrict__ b1,
                     const float* __restrict__ denom1, const float* __restrict__ S1,
                     float* __restrict__ H1) {
    int i = blockIdx.x * blockDim.x + threadIdx.x;
    if (i >= NN * F1) return;
    int n = i >> 8, c = i & 255, h = c >> 5;
    float v = W1[c] * S1[n * NH + h] / (denom1[n * NH + h] + 1e-16f) + b1[c];
    H1[i] = elu(v);
}

// --- layer-2 GEMM: h2[10000,32] = H1[10000,256] @ W2[256,32] + b2, fp32 WMMA
// one wave per 16x16 C tile; 10000 = 625*16 exactly, 32 = 2*16 exactly -> no tails
__global__ void k_gemm(const float* __restrict__ H1, const float* __restrict__ W2,
                       const float* __restrict__ b2, float* __restrict__ h2) {
    const int NTILES = (NN / 16) * 2;                       // 1250
    int wave = blockIdx.x * (blockDim.x >> 5) + (threadIdx.x >> 5);
    if (wave >= NTILES) return;                             // wave-uniform
    const int m0 = (wave >> 1) * 16;
    const int n0 = (wave & 1) * 16;
    const int lane = threadIdx.x & 31;
    const int lr = lane & 15;
    const int hi = lane >> 4;                               // 0: K pair {0,1}, 1: K pair {2,3}
    const int arow = m0 + lr;

    v8f acc = {0.f, 0.f, 0.f, 0.f, 0.f, 0.f, 0.f, 0.f};
    for (int k0 = 0; k0 < F1; k0 += 4) {
        int kk = k0 + (hi << 1);
        v2f A = { H1[arow * F1 + kk], H1[arow * F1 + kk + 1] };
        v2f B = { W2[kk * ND + n0 + lr], W2[(kk + 1) * ND + n0 + lr] };
        acc = __builtin_amdgcn_wmma_f32_16x16x4_f32(
            false, A, false, B, (short)0, acc, false, false);
    }
#pragma unroll
    for (int i = 0; i < 8; ++i) {
        int row = m0 + i + (hi << 3);
        int col = n0 + lr;
        h2[row * ND + col] = acc[i] + b2[col];
    }
}

// --- layer-2 node attention halves
__global__ void k_alpha2(const float* __restrict__ h2, const float* __restrict__ as2w,
                         const float* __restrict__ ad2w, float* as2, float* ad2) {
    int n = blockIdx.x * blockDim.x + threadIdx.x;
    if (n >= NN) return;
    const float* r = h2 + (size_t)n * ND;
    float s = 0.f, d = 0.f;
#pragma unroll
    for (int j = 0; j < ND; ++j) { float v = r[j]; s += v * as2w[j]; d += v * ad2w[j]; }
    as2[n] = s; ad2[n] = d;
}

__global__ void k_edge2_max(const int* __restrict__ src, const int* __restrict__ dst,
                            const float* __restrict__ as2, const float* __restrict__ ad2,
                            unsigned* emax2) {
    int e = blockIdx.x * blockDim.x + threadIdx.x;
    if (e >= NE) return;
    int s = src[e], t = dst[e];
    atomicMax(&emax2[t], enc_f(lrelu(as2[s] + ad2[t])));
}

__global__ void k_edge2_sum(const int* __restrict__ src, const int* __restrict__ dst,
                            const float* __restrict__ as2, const float* __restrict__ ad2,
                            const unsigned* __restrict__ emax2, float* denom2, float* ex2) {
    int e = blockIdx.x * blockDim.x + threadIdx.x;
    if (e >= NE) return;
    int s = src[e], t = dst[e];
    float ex = expf(lrelu(as2[s] + ad2[t]) - dec_f(emax2[t]));
    atomicAdd(&denom2[t], ex);
    ex2[e] = ex;
}

// one thread per (edge, dim): coalesced h2 reads, contiguous 32-lane atomic bursts
__global__ void k_edge2_agg(const int* __restrict__ src, const int* __restrict__ dst,
                            const float* __restrict__ ex2, const float* __restrict__ h2,
                            float* agg2) {
    int i = blockIdx.x * blockDim.x + threadIdx.x;
    if (i >= NE * ND) return;
    int e = i >> 5, d = i & 31;
    int s = src[e], t = dst[e];
    atomicAdd(&agg2[t * ND + d], ex2[e] * h2[s * ND + d]);
}

// --- layer-2 output + elu + mean-pool (LDS partials -> 32 global atomics/block)
__global__ void k_out2(const float* __restrict__ agg2, const float* __restrict__ denom2,
                       const float* __restrict__ b2, float* pooled) {
    __shared__ float part[ND];
    int tid = threadIdx.x;
    if (tid < ND) part[tid] = 0.f;
    __syncthreads();
    int i = blockIdx.x * blockDim.x + tid;
    if (i < NN * ND) {
        int n = i >> 5, d = i & 31;
        float v = elu(agg2[i] / (denom2[n] + 1e-16f) + b2[d]);
        atomicAdd(&part[d], v);
    }
    __syncthreads();
    if (tid < ND) atomicAdd(&pooled[tid], part[tid]);
}

__global__ void k_final(const float* __restrict__ pooled, const float* __restrict__ Wc,
                        const float* __restrict__ bc, float* out) {
    int d = threadIdx.x;
    float v = (d < ND) ? (pooled[d] * (1.0f / NN)) * Wc[d] : 0.f;
    for (int off = 16; off > 0; off >>= 1) v += __shfl_down(v, off, 32);
    if (d == 0) out[0] = v + bc[0];
}

extern "C" void kernel_launch(void* const* d_in, const int* in_sizes, int n_in,
                              void* d_out, int out_size, void* d_ws, size_t ws_size,
                              hipStream_t stream) {
    const float* x    = (const float*)d_in[0];
    const int*   ei   = (const int*)d_in[1];     // int32 (jax default x64-disabled)
    const float* W1   = (const float*)d_in[2];
    const float* as1  = (const float*)d_in[3];
    const float* ad1  = (const float*)d_in[4];
    const float* b1   = (const float*)d_in[5];
    const float* W2   = (const float*)d_in[6];
    const float* as2w = (const float*)d_in[7];
    const float* ad2w = (const float*)d_in[8];
    const float* b2   = (const float*)d_in[9];
    const float* Wc   = (const float*)d_in[10];
    const float* bc   = (const float*)d_in[11];
    float* out = (float*)d_out;
    (void)in_sizes; (void)n_in; (void)out_size; (void)ws_size;

    const int* src = ei;
    const int* dst = ei + NE;

    char* base = (char*)d_ws;
    size_t off = 0;
    auto alloc = [&](size_t bytes) -> char* {
        char* p = base + off;
        off += (bytes + 255) & ~(size_t)255;
        return p;
    };
    float*    cs1    = (float*)alloc(NH * sizeof(float));
    float*    cd1    = (float*)alloc(NH * sizeof(float));
    unsigned* emax1  = (unsigned*)alloc((size_t)NN * NH * 4);
    float*    denom1 = (float*)alloc((size_t)NN * NH * 4);
    float*    S1     = (float*)alloc((size_t)NN * NH * 4);
    float*    H1     = (float*)alloc((size_t)NN * F1 * 4);    // 10.24 MB
    float*    h2     = (float*)alloc((size_t)NN * ND * 4);
    float*    as2    = (float*)alloc((size_t)NN * 4);
    float*    ad2    = (float*)alloc((size_t)NN * 4);
    unsigned* emax2  = (unsigned*)alloc((size_t)NN * 4);
    float*    denom2 = (float*)alloc((size_t)NN * 4);
    float*    ex2    = (float*)alloc((size_t)NE * 4);
    float*    agg2   = (float*)alloc((size_t)NN * ND * 4);
    float*    pooled = (float*)alloc((size_t)ND * 4);

    const int B = 256;
    k_prep<<<1, 32, 0, stream>>>(W1, as1, ad1, cs1, cd1);
    k_init<<<(NN * ND + B - 1) / B, B, 0, stream>>>(emax1, denom1, S1, emax2, denom2, agg2, pooled);
    k_edge1_max<<<(NE + B - 1) / B, B, 0, stream>>>(src, dst, x, cs1, cd1, emax1);
    k_edge1_sum<<<(NE + B - 1) / B, B, 0, stream>>>(src, dst, x, cs1, cd1, emax1, denom1, S1);
    k_h1<<<(NN * F1 + B - 1) / B, B, 0, stream>>>(W1, b1, denom1, S1, H1);
    k_gemm<<<(1250 + 3) / 4, 128, 0, stream>>>(H1, W2, b2, h2);   // 4 waves/block (wave32)
    k_alpha2<<<(NN + B - 1) / B, B, 0, stream>>>(h2, as2w, ad2w, as2, ad2);
    k_edge2_max<<<(NE + B - 1) / B, B, 0, stream>>>(src, dst, as2, ad2, emax2);
    k_edge2_sum<<<(NE + B - 1) / B, B, 0, stream>>>(src, dst, as2, ad2, emax2, denom2, ex2);
    k_edge2_agg<<<(NE * ND + B - 1) / B, B, 0, stream>>>(src, dst, ex2, h2, agg2);
    k_out2<<<(NN * ND + B - 1) / B, B, 0, stream>>>(agg2, denom2, b2, pooled);
    k_final<<<1, 32, 0, stream>>>(pooled, Wc, bc, out);
}